// Model_37108517437743
// MI455X (gfx1250) — compile-verified
//
#include <hip/hip_runtime.h>

// MoE finalize-routing (unpermute + scale + bias + skips), MI455X / gfx1250.
//
// out[n,h] = s0*(exp[map[n],h]        + bias[e0,h])
//          + s1*(exp[map[NROWS+n],h]  + bias[e1,h])
//          + skip1[n,h] + skip2[n,h]
//
// Memory-bound: ~1.6 GB total traffic -> ~70 us floor at 23.3 TB/s.
// Strategy: per-row workgroup, async global->LDS stage of the two gathered
// expert rows (ASYNCcnt) with NT temporal hint, NT hints on the other
// read-once streams, RT (default) on the 1 MB bias table so it stays in L2,
// b128 accesses everywhere.

#define HDIM  4096
#define NROWS 16384
#define KEXP  2
#define TPB   256
#define CHUNKS (HDIM / 4 / TPB)   // 4 float4s per thread

// CPol immediate: TH in bits [2:0] (0=RT, 1=NT, ...), SCOPE in bits [4:3].
#define CPOL_TH_NT 1

typedef float v4f __attribute__((ext_vector_type(4)));
typedef int   v4i __attribute__((ext_vector_type(4)));

#if defined(__gfx1250__) && __has_builtin(__builtin_amdgcn_global_load_async_to_lds_b128)
#define USE_ASYNC 1
#else
#define USE_ASYNC 0
#endif

// AS1 (global) / AS3 (LDS) qualified int4 pointers — exact builtin param types.
typedef __attribute__((address_space(1))) v4i gas_v4i;
typedef __attribute__((address_space(3))) v4i las_v4i;

__device__ __forceinline__ void async_copy_b128_nt(const void* gsrc, void* ldst) {
#if USE_ASYNC
    __builtin_amdgcn_global_load_async_to_lds_b128(
        (gas_v4i*)(void*)gsrc, (las_v4i*)ldst, /*offset=*/0, /*cpol=*/CPOL_TH_NT);
#else
    *(v4f*)ldst = *(const v4f*)gsrc;   // correct fallback (no async path)
#endif
}

__device__ __forceinline__ void wait_async0() {
#if defined(__gfx1250__)
#if __has_builtin(__builtin_amdgcn_s_wait_asynccnt)
    __builtin_amdgcn_s_wait_asynccnt(0);
#else
    asm volatile("s_wait_asynccnt 0" ::: "memory");
#endif
#endif
}

__global__ __launch_bounds__(TPB) void moe_finalize_kernel(
    const float* __restrict__ exp_rows,   // [NROWS*KEXP, HDIM]
    const float* __restrict__ skip1,      // [NROWS, HDIM]
    const float* __restrict__ skip2,      // [NROWS, HDIM]
    const float* __restrict__ bias,       // [E, HDIM]
    const float* __restrict__ scales,     // [NROWS, KEXP]
    const int*   __restrict__ s2d,        // [NROWS*KEXP] inverse permutation
    const int*   __restrict__ expert_id,  // [NROWS, KEXP]
    float*       __restrict__ out)        // [NROWS, HDIM]
{
    __shared__ float lds[KEXP * HDIM];    // 32 KB: the two gathered expert rows

    const int n = blockIdx.x;
    const int t = threadIdx.x;

    // Column arrangement (drop_pad_mode == 0): flat index = k*NROWS + n.
    const int   src0 = s2d[n];
    const int   src1 = s2d[NROWS + n];
    const float s0   = scales[2 * n + 0];
    const float s1   = scales[2 * n + 1];
    const int   e0   = expert_id[2 * n + 0];
    const int   e1   = expert_id[2 * n + 1];

    const v4f* __restrict__ g0 = (const v4f*)(exp_rows + (size_t)src0 * HDIM);
    const v4f* __restrict__ g1 = (const v4f*)(exp_rows + (size_t)src1 * HDIM);

    // Stage both gathered rows into LDS with async NT b128 copies.
    // Thread t owns float4 slots {t, t+256, t+512, t+768} of each row and
    // reads back exactly those slots -> only an ASYNCcnt wait is needed,
    // no workgroup barrier.
#pragma unroll
    for (int c = 0; c < CHUNKS; ++c) {
        const int i = t + c * TPB;
        async_copy_b128_nt(&g0[i], &lds[0]    + (size_t)i * 4);
        async_copy_b128_nt(&g1[i], &lds[HDIM] + (size_t)i * 4);
    }

    const v4f* __restrict__ b0 = (const v4f*)(bias  + (size_t)e0 * HDIM);
    const v4f* __restrict__ b1 = (const v4f*)(bias  + (size_t)e1 * HDIM);
    const v4f* __restrict__ k1 = (const v4f*)(skip1 + (size_t)n  * HDIM);
    const v4f* __restrict__ k2 = (const v4f*)(skip2 + (size_t)n  * HDIM);
    v4f*       __restrict__ ov = (v4f*)      (out   + (size_t)n  * HDIM);
    const v4f* l0 = (const v4f*)&lds[0];
    const v4f* l1 = (const v4f*)&lds[HDIM];

#if USE_ASYNC
    wait_async0();
#endif

#pragma unroll
    for (int c = 0; c < CHUNKS; ++c) {
        const int i = t + c * TPB;
        const v4f a0  = l0[i];                                   // ds_load_b128
        const v4f a1  = l1[i];
        const v4f bb0 = b0[i];                                   // RT: bias stays in L2
        const v4f bb1 = b1[i];
        const v4f sk1 = __builtin_nontemporal_load(&k1[i]);      // NT: read-once streams
        const v4f sk2 = __builtin_nontemporal_load(&k2[i]);
        const v4f r = s0 * (a0 + bb0) + s1 * (a1 + bb1) + (sk1 + sk2);
        __builtin_nontemporal_store(r, &ov[i]);                  // NT store
    }
}

extern "C" void kernel_launch(void* const* d_in, const int* in_sizes, int n_in,
                              void* d_out, int out_size, void* d_ws, size_t ws_size,
                              hipStream_t stream) {
    const float* exp_rows  = (const float*)d_in[0];
    const float* skip1     = (const float*)d_in[1];
    const float* skip2     = (const float*)d_in[2];
    const float* bias      = (const float*)d_in[3];
    const float* scales    = (const float*)d_in[4];
    const int*   s2d       = (const int*)  d_in[5];
    const int*   expert_id = (const int*)  d_in[6];
    // d_in[7] = drop_pad_mode scalar (0 -> column arrangement, handled above)
    float* out = (float*)d_out;

    moe_finalize_kernel<<<NROWS, TPB, 0, stream>>>(
        exp_rows, skip1, skip2, bias, scales, s2d, expert_id, out);
}